// GCN_76982993813545
// MI455X (gfx1250) — compile-verified
//
#include <hip/hip_runtime.h>

typedef __attribute__((ext_vector_type(2))) float v2f;
typedef __attribute__((ext_vector_type(8))) float v8f;

#define NN    9
#define VV    9
#define IN_F  40
#define HID   80
#define OUTF  160
#define SLOPE 0.2f
#define STR   82   // LDS row stride in floats (bank-conflict-free for b64 reads)

__device__ __forceinline__ float leaky(float v) { return v > 0.f ? v : v * SLOPE; }

// A-operand (16xK) read from LDS tile buffer: lane<16 -> (k0,k0+1), lane>=16 -> (k0+2,k0+3)
__device__ __forceinline__ v2f lds_read_a(const float* buf, int lane, int k0) {
  int row = lane & 15;
  int kk  = k0 + ((lane >> 4) << 1);
  v2f a;
  a.x = buf[row * STR + kk];
  a.y = buf[row * STR + kk + 1];
  return a;
}

// B-operand (4x16 slab rows k0..k0+3, cols c0..c0+15) from LDS
__device__ __forceinline__ v2f lds_read_b(const float* buf, int lane, int k0, int c0) {
  int col = c0 + (lane & 15);
  int kk  = k0 + ((lane >> 4) << 1);
  v2f b;
  b.x = buf[kk * STR + col];
  b.y = buf[(kk + 1) * STR + col];
  return b;
}

// B-operand slab streamed directly from global weights (row-major K x ncols)
__device__ __forceinline__ v2f glb_read_b(const float* __restrict__ W, int ncols,
                                          int lane, int k0, int c0) {
  int col = c0 + (lane & 15);
  int kk  = k0 + ((lane >> 4) << 1);
  v2f b;
  b.x = W[kk * ncols + col];
  b.y = W[(kk + 1) * ncols + col];
  return b;
}

#define WMMA_F32(A, B, C) \
  __builtin_amdgcn_wmma_f32_16x16x4_f32(false, (A), false, (B), (short)0, (C), false, false)

__global__ __launch_bounds__(32) void gcn_kernel(const int* __restrict__ x,
                                                 const float* __restrict__ emb,
                                                 const float* __restrict__ W1,
                                                 const float* __restrict__ W2,
                                                 const float* __restrict__ A,
                                                 float* __restrict__ out) {
  __shared__ float buf0[16 * STR];  // Y, later H
  __shared__ float buf1[16 * STR];  // Z1, later Z2
  const int lane = threadIdx.x;
  const int p    = blockIdx.x;

  const float* embp = emb + (size_t)p * VV * IN_F;
  const float* W1p  = W1 + (size_t)p * IN_F * HID;
  const float* W2p  = W2 + (size_t)p * HID * OUTF;
  const float* Ap   = A + (size_t)p * NN * NN;
  float* outp       = out + (size_t)p * NN * OUTF;

  // ---- zero buf0 (padding rows 9..15 and cols 40..47 must be 0) ----
  for (int i = lane; i < 16 * STR; i += 32) buf0[i] = 0.f;

  // ---- gather Y[n][f] = emb[x[n]][f] into buf0 ----
  for (int i = lane; i < NN * IN_F; i += 32) {
    int n = i / IN_F;
    int f = i - n * IN_F;
    buf0[n * STR + f] = embp[x[n] * IN_F + f];
  }

  // ---- A_pad (16x16, zeros outside 9x9) as register A-operand, K=16 -> 4 k-steps ----
  v2f a_op[4];
  {
    int row   = lane & 15;
    int kbase = (lane >> 4) << 1;
#pragma unroll
    for (int ks = 0; ks < 4; ++ks) {
      int k      = ks * 4 + kbase;
      a_op[ks].x = (row < NN && k < NN) ? Ap[row * NN + k] : 0.f;
      a_op[ks].y = (row < NN && k + 1 < NN) ? Ap[row * NN + k + 1] : 0.f;
    }
  }

  asm volatile("s_wait_dscnt 0" ::: "memory");

  // ---- Z1 = A_pad @ Y : 16x48 (3 col tiles), K=16 ----
  v8f z1[3];
#pragma unroll
  for (int t = 0; t < 3; ++t) {
    v8f acc = {};
#pragma unroll
    for (int ks = 0; ks < 4; ++ks) {
      v2f b = lds_read_b(buf0, lane, ks * 4, t * 16);
      acc   = WMMA_F32(a_op[ks], b, acc);
    }
    z1[t] = acc;
  }
  // C/D layout -> LDS (row = vgpr + (lane>=16 ? 8 : 0), col = lane&15)
  {
    int col = lane & 15;
    int r0  = (lane >> 4) << 3;
#pragma unroll
    for (int t = 0; t < 3; ++t)
#pragma unroll
      for (int r = 0; r < 8; ++r) buf1[(r0 + r) * STR + t * 16 + col] = z1[t][r];
  }
  asm volatile("s_wait_dscnt 0" ::: "memory");

  // ---- T1 = Z1 @ W1 : 16x80, K=40 ; H = leaky(T1) -> buf0 ----
  v8f h[5];
#pragma unroll
  for (int t = 0; t < 5; ++t) { v8f z = {}; h[t] = z; }
  for (int k0 = 0; k0 < IN_F; k0 += 4) {
    v2f a = lds_read_a(buf1, lane, k0);
#pragma unroll
    for (int t = 0; t < 5; ++t) {
      v2f b = glb_read_b(W1p, HID, lane, k0, t * 16);
      h[t]  = WMMA_F32(a, b, h[t]);
    }
  }
  {
    int col = lane & 15;
    int r0  = (lane >> 4) << 3;
#pragma unroll
    for (int t = 0; t < 5; ++t)
#pragma unroll
      for (int r = 0; r < 8; ++r) buf0[(r0 + r) * STR + t * 16 + col] = leaky(h[t][r]);
  }
  asm volatile("s_wait_dscnt 0" ::: "memory");

  // ---- Z2 = A_pad @ H : 16x80, K=16 -> buf1 ----
  v8f z2[5];
#pragma unroll
  for (int t = 0; t < 5; ++t) {
    v8f acc = {};
#pragma unroll
    for (int ks = 0; ks < 4; ++ks) {
      v2f b = lds_read_b(buf0, lane, ks * 4, t * 16);
      acc   = WMMA_F32(a_op[ks], b, acc);
    }
    z2[t] = acc;
  }
  {
    int col = lane & 15;
    int r0  = (lane >> 4) << 3;
#pragma unroll
    for (int t = 0; t < 5; ++t)
#pragma unroll
      for (int r = 0; r < 8; ++r) buf1[(r0 + r) * STR + t * 16 + col] = z2[t][r];
  }
  asm volatile("s_wait_dscnt 0" ::: "memory");

  // ---- T2 = Z2 @ W2 : 16x160, K=80, 10 col tiles in 2 chunks of 5 ; leaky ; store ----
  for (int chunk = 0; chunk < 2; ++chunk) {
    v8f acc[5];
#pragma unroll
    for (int t = 0; t < 5; ++t) { v8f z = {}; acc[t] = z; }
    for (int k0 = 0; k0 < HID; k0 += 4) {
      v2f a = lds_read_a(buf1, lane, k0);
#pragma unroll
      for (int t = 0; t < 5; ++t) {
        v2f b  = glb_read_b(W2p, OUTF, lane, k0, (chunk * 5 + t) * 16);
        acc[t] = WMMA_F32(a, b, acc[t]);
      }
    }
    int col = lane & 15;
    int r0  = (lane >> 4) << 3;
#pragma unroll
    for (int t = 0; t < 5; ++t) {
      int c = (chunk * 5 + t) * 16 + col;
#pragma unroll
      for (int r = 0; r < 8; ++r) {
        int row = r0 + r;
        if (row < NN) outp[row * OUTF + c] = leaky(acc[t][r]);
      }
    }
  }
}

extern "C" void kernel_launch(void* const* d_in, const int* in_sizes, int n_in,
                              void* d_out, int out_size, void* d_ws, size_t ws_size,
                              hipStream_t stream) {
  const int* x     = (const int*)d_in[0];
  const float* emb = (const float*)d_in[1];
  const float* W1  = (const float*)d_in[2];
  const float* W2  = (const float*)d_in[3];
  const float* A   = (const float*)d_in[4];
  float* out       = (float*)d_out;
  const int P      = in_sizes[4] / (NN * NN);  // 2048
  gcn_kernel<<<P, 32, 0, stream>>>(x, emb, W1, W2, A, out);
}